// RegionProposal_2439541424356
// MI455X (gfx1250) — compile-verified
//
#include <hip/hip_runtime.h>
#include <math.h>

#define NMS_THREADS   1024
#define NUM_POST_NMS  300
#define IOU_THR       0.7f

__device__ __forceinline__ float neg_inf_f() { return __uint_as_float(0xff800000u); }

// ---------------------------------------------------------------------------
// Kernel A: decode deltas against anchors, clip to image, copy scores to a
// read-write workspace array. Pure streaming, float4 (b128) loads/stores.
// TARGET_STDS = 1, TARGET_MEANS = 0 -> identity, folded away.
// ---------------------------------------------------------------------------
__global__ __launch_bounds__(256) void decode_clip_kernel(
    const float4* __restrict__ deltas, const float4* __restrict__ anchors,
    const float*  __restrict__ scores_in, const float* __restrict__ image_shape,
    float4* __restrict__ boxes, float* __restrict__ scores_rw, int n)
{
    int i = blockIdx.x * blockDim.x + threadIdx.x;
    if (i >= n) return;
    float4 a = anchors[i];
    float4 d = deltas[i];
    float h  = a.z - a.x;
    float w  = a.w - a.y;
    float cy = a.x + 0.5f * h;
    float cx = a.y + 0.5f * w;
    float ncy = d.x * h + cy;
    float ncx = d.y * w + cx;
    float nh  = expf(d.z) * h;
    float nw  = expf(d.w) * w;
    float maxh = image_shape[0];
    float maxw = image_shape[1];
    float4 b;
    b.x = fminf(fmaxf(ncy - 0.5f * nh, 0.f), maxh);
    b.y = fminf(fmaxf(ncx - 0.5f * nw, 0.f), maxw);
    b.z = fminf(fmaxf(ncy + 0.5f * nh, 0.f), maxh);
    b.w = fminf(fmaxf(ncx + 0.5f * nw, 0.f), maxw);
    boxes[i]     = b;
    scores_rw[i] = scores_in[i];
}

// ---------------------------------------------------------------------------
// Kernel B: persistent single-workgroup greedy NMS (300 serial selections).
// Each selection = fused (suppress-by-previous-winner + block argmax) scan.
// Box tiles are staged global->LDS with the CDNA5 async-copy engine
// (ASYNCcnt), double buffered. Each wave stages only its own lanes' slots,
// so ASYNCcnt ordering is wave-local; the wait level is made wave-uniform
// and matches exactly what the wave issued (fixes partial-final-chunk waves).
// ---------------------------------------------------------------------------
__global__ __launch_bounds__(NMS_THREADS) void nms_kernel(
    const float4* __restrict__ boxes, float* __restrict__ scores,
    float4* __restrict__ out, int n)
{
    __shared__ float4 stage[2][NMS_THREADS];   // 32 KB double buffer
    __shared__ float  red_s[NMS_THREADS / 32];
    __shared__ int    red_i[NMS_THREADS / 32];
    __shared__ float  bc_s;
    __shared__ int    bc_i;

    const int t         = threadIdx.x;
    const int lane      = t & 31;
    const int wave      = t >> 5;
    const int wave_base = wave << 5;           // first lane's thread index
    const int nchunk    = (n + NMS_THREADS - 1) / NMS_THREADS;
    const float NEG     = neg_inf_f();

    bool   have_prev = false;
    int    prev_idx  = -1;
    float4 pb        = make_float4(0.f, 0.f, 0.f, 0.f);
    float  parea     = 0.f;

    for (int sel = 0; sel < NUM_POST_NMS; ++sel) {
        // Prologue: async-stage chunk 0 into buffer 0 (n > NMS_THREADS, so
        // every wave has valid lanes here).
        {
            int i0 = t;
            if (i0 < n) {
                unsigned      ldsoff = (unsigned)(size_t)(&stage[0][t]);
                const float4* gp     = boxes + i0;
                asm volatile("global_load_async_to_lds_b128 %0, %1, off"
                             :: "v"(ldsoff), "v"(gp) : "memory");
            }
        }

        float bs = NEG;
        int   bi = 0x7fffffff;

        for (int k = 0; k < nchunk; ++k) {
            const int buf = k & 1;
            if (k + 1 < nchunk) {
                // Prefetch next chunk into the other buffer. wave_issues is
                // wave-uniform: lanes are ascending, so if the first lane is
                // OOB the whole wave issued nothing and must drain to 0.
                int  i1          = (k + 1) * NMS_THREADS + t;
                bool wave_issues = ((k + 1) * NMS_THREADS + wave_base) < n;
                if (i1 < n) {
                    unsigned      ldsoff = (unsigned)(size_t)(&stage[buf ^ 1][t]);
                    const float4* gp     = boxes + i1;
                    asm volatile("global_load_async_to_lds_b128 %0, %1, off"
                                 :: "v"(ldsoff), "v"(gp) : "memory");
                }
                if (wave_issues) {
                    asm volatile("s_wait_asynccnt 1" ::: "memory");
                } else {
                    asm volatile("s_wait_asynccnt 0" ::: "memory");
                }
            } else {
                asm volatile("s_wait_asynccnt 0" ::: "memory");
            }

            int i = k * NMS_THREADS + t;
            if (i < n) {
                float s = scores[i];
                if (have_prev && s != NEG) {
                    float4 b    = stage[buf][t];
                    float yy1   = fmaxf(pb.x, b.x);
                    float xx1   = fmaxf(pb.y, b.y);
                    float yy2   = fminf(pb.z, b.z);
                    float xx2   = fminf(pb.w, b.w);
                    float inter = fmaxf(yy2 - yy1, 0.f) * fmaxf(xx2 - xx1, 0.f);
                    float barea = (b.z - b.x) * (b.w - b.y);
                    float iou   = inter / fmaxf(parea + barea - inter, 1e-8f);
                    if (iou > IOU_THR || i == prev_idx) {  // i==prev_idx: .at[idx].set(-inf)
                        s = NEG;
                        scores[i] = s;
                    }
                }
                if (s > bs || (s == bs && i < bi)) { bs = s; bi = i; }
            }
            // Retire this wave's LDS reads before the async engine can
            // overwrite this buffer on a later iteration (WAR fence).
            asm volatile("s_wait_dscnt 0" ::: "memory");
        }

        // Wave32 argmax reduce (lowest index wins ties, matching jnp.argmax).
        for (int off = 16; off > 0; off >>= 1) {
            float os = __shfl_xor(bs, off, 32);
            int   oi = __shfl_xor(bi, off, 32);
            if (os > bs || (os == bs && oi < bi)) { bs = os; bi = oi; }
        }
        if (lane == 0) { red_s[wave] = bs; red_i[wave] = bi; }
        __syncthreads();
        if (wave == 0) {
            bs = red_s[lane];
            bi = red_i[lane];
            for (int off = 16; off > 0; off >>= 1) {
                float os = __shfl_xor(bs, off, 32);
                int   oi = __shfl_xor(bi, off, 32);
                if (os > bs || (os == bs && oi < bi)) { bs = os; bi = oi; }
            }
            if (lane == 0) { bc_s = bs; bc_i = bi; }
        }
        __syncthreads();

        float best_s = bc_s;
        int   best_i = bc_i;
        bool  valid  = (best_s != NEG) && (best_i < n);

        float4 bb = make_float4(0.f, 0.f, 0.f, 0.f);
        if (valid) bb = boxes[best_i];     // broadcast load, L2-hot
        if (t == 0) out[sel] = bb;         // invalid -> zeros (matches where())

        have_prev = valid;                 // !valid => scores untouched next pass
        prev_idx  = best_i;
        pb        = bb;
        parea     = (bb.z - bb.x) * (bb.w - bb.y);
        __syncthreads();                   // protect red_/bc_ reuse
    }
}

// ---------------------------------------------------------------------------
extern "C" void kernel_launch(void* const* d_in, const int* in_sizes, int n_in,
                              void* d_out, int out_size, void* d_ws, size_t ws_size,
                              hipStream_t stream)
{
    const float4* deltas  = (const float4*)d_in[0];
    const float4* anchors = (const float4*)d_in[1];
    const float*  scores  = (const float*)d_in[2];
    const float*  imshape = (const float*)d_in[3];
    const int n = in_sizes[2];                      // scores element count

    float4* boxes    = (float4*)d_ws;
    float*  scoresrw = (float*)((char*)d_ws + (size_t)n * sizeof(float4));
    float4* out      = (float4*)d_out;              // 300 x float4

    decode_clip_kernel<<<(n + 255) / 256, 256, 0, stream>>>(
        deltas, anchors, scores, imshape, boxes, scoresrw, n);
    nms_kernel<<<1, NMS_THREADS, 0, stream>>>(boxes, scoresrw, out, n);
}